// ConvolutionLayer_17437567222234
// MI455X (gfx1250) — compile-verified
//
#include <hip/hip_runtime.h>

// Reference: h = X @ W ; out = relu(segment_sum(h[src]*val -> dst))
// X: [N,512] f32, W: [512,256] f32, edge_vals f32, edge_src/dst int64.
// Phase 1: WMMA f32 GEMM (V_WMMA_F32_16X16X4_F32), LDS double-buffered via
//          GLOBAL_LOAD_ASYNC_TO_LDS_B128 (ASYNCcnt) on full tiles.
// Phase 2: zero out, scatter-add with global_atomic_add_f32 (L2-resident).
// Phase 3: in-place ReLU.

#define D_IN  512
#define D_OUT 256

typedef __attribute__((ext_vector_type(2))) float v2f;
typedef __attribute__((ext_vector_type(8))) float v8f;
typedef __attribute__((__vector_size__(4 * sizeof(int)))) int v4i;

// pointer-to-AS1-int4 and pointer-to-AS3-int4 (per builtin signature from
// round-2 diagnostic: (v4i AS1*, v4i AS3*, imm offset, imm cpol))
typedef __attribute__((address_space(1))) v4i* v4i_g;
typedef __attribute__((address_space(3))) v4i* v4i_l;

#define M_BLK 128
#define N_BLK 64
#define K_BLK 16
#define A_PAD 20   // A row stride (floats): 80B -> every c4*4 dest 16B-aligned, conflict-free b64 reads
#define B_PAD 72   // B row stride (floats): 288B -> 16B-aligned dests; +16 banks per 2 rows

#define ASYNC_CP_B128(gp, lp) \
    __builtin_amdgcn_global_load_async_to_lds_b128((v4i_g)(gp), (v4i_l)(lp), 0, 0)

#if __has_builtin(__builtin_amdgcn_s_wait_asynccnt)
#define WAIT_ASYNC0() __builtin_amdgcn_s_wait_asynccnt(0)
#else
#define WAIT_ASYNC0() asm volatile("s_wait_asynccnt 0x0" ::: "memory")
#endif

__global__ __launch_bounds__(256) void gemm_xw_wmma(
    const float* __restrict__ X, const float* __restrict__ W,
    float* __restrict__ H, int nNodes)
{
    __shared__ float As[2][M_BLK * A_PAD];   // 2 x 128x16 (+pad) = 20.0 KB
    __shared__ float Bs[2][K_BLK * B_PAD];   // 2 x 16x64  (+pad) =  9.0 KB

    const int tid    = threadIdx.x;
    const int wave   = tid >> 5;
    const int lane   = tid & 31;
    const int lane15 = lane & 15;
    const int hiHalf = lane >> 4;            // 0: lanes 0-15, 1: lanes 16-31
    const int m0     = blockIdx.x * M_BLK;
    const int n0     = blockIdx.y * N_BLK;
    const int mW     = (wave >> 1) * 32;     // wave M offset (4 waves in M)
    const int nW     = (wave & 1) * 32;      // wave N offset (2 waves in N)

    // staging slot assignment (A: 2 x float4 per thread, B: 1 x float4)
    const int bRow = tid >> 4;               // 0..15
    const int bC   = (tid & 15) * 4;         // 0..60

    v8f acc[2][2] = {};

    auto compute = [&](int buf) {
        #pragma unroll
        for (int kk = 0; kk < K_BLK; kk += 4) {
            const int k0 = kk + hiHalf * 2;  // this half-wave's K pair {k0, k0+1}
            v2f a[2], b[2];
            #pragma unroll
            for (int sm = 0; sm < 2; ++sm) {
                const float* p = &As[buf][(mW + sm * 16 + lane15) * A_PAD + k0];
                a[sm].x = p[0];              // -> VGPR0 (K even)
                a[sm].y = p[1];              // -> VGPR1 (K odd)
            }
            #pragma unroll
            for (int sn = 0; sn < 2; ++sn) {
                const int n = nW + sn * 16 + lane15;
                b[sn].x = Bs[buf][k0 * B_PAD + n];
                b[sn].y = Bs[buf][(k0 + 1) * B_PAD + n];
            }
            #pragma unroll
            for (int sm = 0; sm < 2; ++sm)
                #pragma unroll
                for (int sn = 0; sn < 2; ++sn)
                    acc[sm][sn] = __builtin_amdgcn_wmma_f32_16x16x4_f32(
                        false, a[sm], false, b[sn],
                        (short)0, acc[sm][sn], false, false);
        }
    };

    const bool fullTile = (m0 + M_BLK) <= nNodes;

    if (fullTile) {
        // ---- async double-buffered pipeline: copy(next) overlaps compute(cur) ----
        auto issue = [&](int buf, int k) {
            #pragma unroll
            for (int i = 0; i < 2; ++i) {
                int slot = tid * 2 + i;
                int row  = slot >> 2;        // 0..127
                int c4   = (slot & 3) * 4;   // 0,4,8,12
                ASYNC_CP_B128(&X[(size_t)(m0 + row) * D_IN + k + c4],
                              &As[buf][row * A_PAD + c4]);
            }
            ASYNC_CP_B128(&W[(size_t)(k + bRow) * D_OUT + n0 + bC],
                          &Bs[buf][bRow * B_PAD + bC]);
        };

        issue(0, 0);
        WAIT_ASYNC0();
        __syncthreads();
        for (int k = 0; k < D_IN; k += K_BLK) {
            const int cur = (k >> 4) & 1;
            if (k + K_BLK < D_IN) issue(cur ^ 1, k + K_BLK);
            compute(cur);
            WAIT_ASYNC0();       // my writes into the other buffer are done
            __syncthreads();     // everyone done reading cur + sees next buffer
        }
    } else {
        // ---- tail block (partial rows): guarded synchronous staging ----
        for (int k = 0; k < D_IN; k += K_BLK) {
            #pragma unroll
            for (int i = 0; i < 2; ++i) {
                int slot = tid * 2 + i;
                int row  = slot >> 2;
                int c4   = (slot & 3) * 4;
                float4 v = make_float4(0.f, 0.f, 0.f, 0.f);
                if (m0 + row < nNodes)
                    v = *(const float4*)&X[(size_t)(m0 + row) * D_IN + k + c4];
                float* d = &As[0][row * A_PAD + c4];
                d[0] = v.x; d[1] = v.y; d[2] = v.z; d[3] = v.w;
            }
            {
                float4 v = *(const float4*)&W[(size_t)(k + bRow) * D_OUT + n0 + bC];
                float* d = &Bs[0][bRow * B_PAD + bC];
                d[0] = v.x; d[1] = v.y; d[2] = v.z; d[3] = v.w;
            }
            __syncthreads();
            compute(0);
            __syncthreads();
        }
    }

    // ---- store 32x32 per wave; C layout: VGPR r -> M=r (lo half) / M=8+r (hi half)
    #pragma unroll
    for (int sm = 0; sm < 2; ++sm) {
        #pragma unroll
        for (int r = 0; r < 8; ++r) {
            int row = m0 + mW + sm * 16 + hiHalf * 8 + r;
            if (row < nNodes) {
                #pragma unroll
                for (int sn = 0; sn < 2; ++sn)
                    H[(size_t)row * D_OUT + n0 + nW + sn * 16 + lane15] = acc[sm][sn][r];
            }
        }
    }
}

__global__ __launch_bounds__(256) void zero_f4(float4* __restrict__ p, int n4)
{
    int i = blockIdx.x * blockDim.x + threadIdx.x;
    if (i < n4) p[i] = make_float4(0.f, 0.f, 0.f, 0.f);
}

// One edge per 64-thread group (2 waves); edge metadata is wave-uniform, so
// force it into SGPRs with readfirstlane -> s_load path for src/dst/val.
__global__ __launch_bounds__(256) void edge_scatter(
    const float* __restrict__ H, const float* __restrict__ vals,
    const long long* __restrict__ esrc, const long long* __restrict__ edst,
    float* __restrict__ out, int nEdges)
{
    int g = blockIdx.x * blockDim.x + threadIdx.x;
    int e = __builtin_amdgcn_readfirstlane(g >> 6);   // uniform within each wave
    int t = (g & 63) * 4;
    if (e >= nEdges) return;
    long long s = esrc[e];
    long long d = edst[e];
    float v = vals[e];
    float4 hv = *(const float4*)&H[(size_t)s * D_OUT + t];
    float* o = &out[(size_t)d * D_OUT + t];
    atomicAdd(o + 0, hv.x * v);
    atomicAdd(o + 1, hv.y * v);
    atomicAdd(o + 2, hv.z * v);
    atomicAdd(o + 3, hv.w * v);
}

__global__ __launch_bounds__(256) void relu_f4(float4* __restrict__ p, int n4)
{
    int i = blockIdx.x * blockDim.x + threadIdx.x;
    if (i < n4) {
        float4 v = p[i];
        v.x = fmaxf(v.x, 0.f);
        v.y = fmaxf(v.y, 0.f);
        v.z = fmaxf(v.z, 0.f);
        v.w = fmaxf(v.w, 0.f);
        p[i] = v;
    }
}

extern "C" void kernel_launch(void* const* d_in, const int* in_sizes, int n_in,
                              void* d_out, int out_size, void* d_ws, size_t ws_size,
                              hipStream_t stream)
{
    const float*     X    = (const float*)d_in[0];
    const float*     W    = (const float*)d_in[1];
    const float*     vals = (const float*)d_in[2];
    const long long* esrc = (const long long*)d_in[3];
    const long long* edst = (const long long*)d_in[4];
    float* out = (float*)d_out;
    float* H   = (float*)d_ws;                 // 50000*256*4 = 51.2 MB scratch

    const int nNodes = in_sizes[0] / D_IN;
    const int nEdges = in_sizes[2];

    // Phase 1: h = X @ W (WMMA fp32, async-LDS double buffering)
    dim3 gGrid((nNodes + M_BLK - 1) / M_BLK, D_OUT / N_BLK);
    gemm_xw_wmma<<<gGrid, 256, 0, stream>>>(X, W, H, nNodes);

    // Phase 2a: zero the accumulator output
    int n4 = out_size / 4;
    zero_f4<<<(n4 + 255) / 256, 256, 0, stream>>>((float4*)out, n4);

    // Phase 2b: scatter-add edges (L2-resident atomics)
    long long thr = (long long)nEdges * 64;
    edge_scatter<<<(unsigned)((thr + 255) / 256), 256, 0, stream>>>(
        H, vals, esrc, edst, out, nEdges);

    // Phase 3: ReLU in place
    relu_f4<<<(n4 + 255) / 256, 256, 0, stream>>>((float4*)out, n4);
}